// MeanAggregator_72000831750328
// MI455X (gfx1250) — compile-verified
//
#include <hip/hip_runtime.h>
#include <hip/hip_bf16.h>

typedef __attribute__((ext_vector_type(2))) float v2f;
typedef __attribute__((ext_vector_type(4))) float v4f;
typedef __attribute__((ext_vector_type(8))) float v8f;

#define F_DIM 128
#define E_DIM 128
#define S_NEIGH 32
#define KSTEPS (F_DIM / 4)   // 32 WMMA K-steps
#define ETILES (E_DIM / 16)  // 8 output column tiles

// ---------------------------------------------------------------------------
// Kernel 1: embed = relu(metric @ W + b) via V_WMMA_F32_16X16X4_F32.
// Block = 128 threads = 4 waves; each wave produces a 16x128 strip of embed.
//
// W is staged into LDS *pre-swizzled into B-fragment order*:
//   sBfrag[(ks*ETILES + e)*32 + lane] = { W[4ks + (lane>>4)*2    ][e*16 + (lane&15)],
//                                         W[4ks + (lane>>4)*2 + 1][e*16 + (lane&15)] }
// so the hot loop fetches each B fragment with ONE aligned ds_load_b64
// straight into an even VGPR pair (no mov-packing, no bank conflicts:
// consecutive lanes hit consecutive 8B). Exactly 64 KB of LDS.
//
// Fragment layouts per CDNA5 ISA 7.12.2:
//   A (16x4 f32):  lane l holds metric[m0+(l&15)][k0+(l>>4)*2 + {0,1}]  (b64 load)
//   C/D (16x16):   VGPR r, lane l -> row m0 + r + (l>>4)*8, col e*16 + (l&15)
// ---------------------------------------------------------------------------
__global__ void __launch_bounds__(128)
embed_gemm_relu_kernel(const float* __restrict__ metric,
                       const float* __restrict__ W,
                       const float* __restrict__ bias,
                       float* __restrict__ embed,
                       int M) {
    __shared__ v2f sBfrag[KSTEPS * ETILES * 32];   // 64 KB

    const int tid = threadIdx.x;
    // Stage + swizzle W: coalesced global reads, scattered one-time LDS writes.
    for (int idx = tid; idx < F_DIM * E_DIM; idx += 128) {
        const int r = idx >> 7;        // W row (k index)
        const int c = idx & 127;       // W col (e index)
        const float val = W[idx];      // coalesced b32
        const int ks   = r >> 2;       // K-step
        const int rj   = r & 3;
        const int j    = rj & 1;       // which half of the v2f
        const int half = rj >> 1;      // lane half (K pair select)
        const int dlane = half * 16 + (c & 15);
        const int e    = c >> 4;
        ((float*)&sBfrag[(ks * ETILES + e) * 32 + dlane])[j] = val;
    }
    __syncthreads();

    const int wave = tid >> 5;
    const int lane = tid & 31;
    const int m0   = (blockIdx.x * 4 + wave) * 16;
    if (m0 >= M) return;               // uniform per wave; EXEC stays all-1s

    const int half = lane >> 4;
    const int l15  = lane & 15;
    const float* __restrict__ arow = metric + (size_t)(m0 + l15) * F_DIM;
    const v2f*   __restrict__ brow = sBfrag + lane;

    v8f acc[ETILES];
#pragma unroll
    for (int e = 0; e < ETILES; ++e) acc[e] = (v8f)0.0f;

    for (int ks = 0; ks < KSTEPS; ++ks) {
        const int kr = ks * 4 + half * 2;
        v2f a;
        a.x = arow[kr + 0];            // one global_load_b64
        a.y = arow[kr + 1];
        const v2f* bf = brow + ks * ETILES * 32;
#pragma unroll
        for (int e = 0; e < ETILES; ++e) {
            const v2f b = bf[e * 32];  // one ds_load_b64, offset e*256
            acc[e] = __builtin_amdgcn_wmma_f32_16x16x4_f32(
                false, a, false, b, (short)0, acc[e], false, false);
        }
    }

    // Epilogue: bias + ReLU + store.
    const int rbase = half * 8;
#pragma unroll
    for (int e = 0; e < ETILES; ++e) {
        const int col = e * 16 + l15;
        const float bv = bias[col];
#pragma unroll
        for (int r = 0; r < 8; ++r) {
            float v = acc[e][r] + bv;
            v = fmaxf(v, 0.0f);
            embed[(size_t)(m0 + rbase + r) * E_DIM + col] = v;
        }
    }
}

// ---------------------------------------------------------------------------
// Kernel 2: out[n] = relu(mean_s embed[idx[n][s]]). One wave per node:
// lane s holds neighbor id s (S==32==wave32), broadcast via __shfl; each lane
// accumulates a float4 slice (32 lanes x 4 = 128 = E). embed (25.6 MB) is
// L2-resident (192 MB), so the ~1 GB of gather reads run at L2 speed; unroll
// by 8 keeps eight global_load_b128 in flight per lane to cover L2 latency.
// ---------------------------------------------------------------------------
__global__ void __launch_bounds__(256)
gather_mean_relu_kernel(const float* __restrict__ embed,
                        const int* __restrict__ neigh_idx,
                        float* __restrict__ out,
                        int N) {
    const int wave = threadIdx.x >> 5;
    const int lane = threadIdx.x & 31;
    const int n = blockIdx.x * (blockDim.x >> 5) + wave;
    if (n >= N) return;

    const int my_nid = neigh_idx[(size_t)n * S_NEIGH + lane];

    v4f sum = (v4f)0.0f;
#pragma unroll 8
    for (int s = 0; s < S_NEIGH; ++s) {
        const int nid = __shfl(my_nid, s, 32);
        const v4f row = ((const v4f*)(embed + (size_t)nid * E_DIM))[lane];
        sum += row;
    }

    sum *= (1.0f / (float)S_NEIGH);
    v4f r;
    r.x = fmaxf(sum.x, 0.0f);
    r.y = fmaxf(sum.y, 0.0f);
    r.z = fmaxf(sum.z, 0.0f);
    r.w = fmaxf(sum.w, 0.0f);
    ((v4f*)(out + (size_t)n * E_DIM))[lane] = r;
}

extern "C" void kernel_launch(void* const* d_in, const int* in_sizes, int n_in,
                              void* d_out, int out_size, void* d_ws, size_t ws_size,
                              hipStream_t stream) {
    const float* metric    = (const float*)d_in[0];
    const float* W         = (const float*)d_in[1];
    const float* bias      = (const float*)d_in[2];
    const int*   neigh_idx = (const int*)d_in[3];
    float*       out       = (float*)d_out;
    float*       embed     = (float*)d_ws;   // M*E*4 = 25.6 MB scratch

    const int M = in_sizes[0] / F_DIM;       // 50000
    const int N = in_sizes[3] / S_NEIGH;     // 65536

    const int mtiles  = (M + 15) / 16;       // 3125 (exact)
    const int blocks1 = (mtiles + 3) / 4;    // 4 waves per block
    embed_gemm_relu_kernel<<<blocks1, 128, 0, stream>>>(metric, W, bias, embed, M);

    const int blocks2 = (N + 7) / 8;         // 8 nodes per 256-thread block
    gather_mean_relu_kernel<<<blocks2, 256, 0, stream>>>(embed, neigh_idx, out, N);
}